// GraphAttentionLayer_51299089384084
// MI455X (gfx1250) — compile-verified
//
#include <hip/hip_runtime.h>
#include <hip/hip_bf16.h>
#include <math.h>

#define IN_F   128
#define OUT_F  32
#define H      4
#define HF     128   // H*OUT_F
#define NEG_SLOPE 0.2f

typedef __attribute__((ext_vector_type(2))) float v2f;
typedef __attribute__((ext_vector_type(8))) float v8f;

// ---------------------------------------------------------------------------
// Init kernels
// ---------------------------------------------------------------------------
__global__ void gat_init_nodes(float* __restrict__ nmax, float* __restrict__ nsum, int NH) {
    int t = blockIdx.x * blockDim.x + threadIdx.x;
    if (t < NH) { nmax[t] = -INFINITY; nsum[t] = 0.0f; }
}

__global__ void gat_init_out(float* __restrict__ out, const float* __restrict__ bias, int NO) {
    int t = blockIdx.x * blockDim.x + threadIdx.x;
    if (t < NO) out[t] = bias[t & (HF - 1)];
}

// ---------------------------------------------------------------------------
// Wh = x @ W  via fp32 WMMA (16x16x4).
// x: (N,128) row-major, W: (H,128,32) -> logical B (128 x 128), col c = h*32+o
// Block: 256 threads = 8 waves. Block tile: 16 rows x 128 cols.
// Wave w computes the 16x16 tile at columns [w*16, w*16+16).
// ---------------------------------------------------------------------------
__global__ void gat_wh_gemm(const float* __restrict__ x,
                            const float* __restrict__ W,
                            float* __restrict__ Wh, int N) {
    const int lane = threadIdx.x & 31;
    const int wave = threadIdx.x >> 5;       // 0..7 -> column tile
    const int m0   = blockIdx.x * 16;
    const int n0   = wave * 16;
    const int h    = n0 >> 5;                // head of this column tile
    const int o0   = n0 & 31;                // col offset inside head
    const int half = lane >> 4;              // 0: K pair {0,1}, 1: K pair {2,3}
    const int l16  = lane & 15;

    int row = m0 + l16;
    if (row >= N) row = N - 1;               // clamp (N % 16 == 0 normally)

    const float* xrow  = x + (size_t)row * IN_F + half * 2;
    // B element: W[h][k][o] at W[h*IN_F*OUT_F + k*OUT_F + o]
    const float* wbase = W + (size_t)h * IN_F * OUT_F + o0 + l16;

    v8f c = {};
#pragma unroll
    for (int k0 = 0; k0 < IN_F; k0 += 4) {
        v2f a;                                // A 16x4: lanes 0-15 K={k0,k0+1}, 16-31 K={k0+2,k0+3}
        a.x = xrow[k0 + 0];
        a.y = xrow[k0 + 1];
        v2f b;                                // B 4x16: mirrored K striping across lane halves
        const int kb = k0 + half * 2;
        b.x = wbase[(size_t)kb * OUT_F];
        b.y = wbase[(size_t)(kb + 1) * OUT_F];
        c = __builtin_amdgcn_wmma_f32_16x16x4_f32(
                /*neg_a=*/false, a, /*neg_b=*/false, b,
                /*c_mod=*/(short)0, c, /*reuse_a=*/false, /*reuse_b=*/false);
    }

    // C/D layout: VGPR r -> M = r (lanes 0-15) or 8+r (lanes 16-31), N = lane%16
    const int col   = n0 + l16;
    const int rbase = m0 + half * 8;
#pragma unroll
    for (int r = 0; r < 8; ++r) {
        int rr = rbase + r;
        if (rr < N) Wh[(size_t)rr * HF + col] = c[r];
    }
}

// ---------------------------------------------------------------------------
// e_l[n,h] = Wh[n, h*32 : h*32+32] . a_l[h];  same for e_r
// ---------------------------------------------------------------------------
__global__ void gat_elr(const float* __restrict__ Wh,
                        const float* __restrict__ a_l,
                        const float* __restrict__ a_r,
                        float* __restrict__ el, float* __restrict__ er, int N) {
    int t = blockIdx.x * blockDim.x + threadIdx.x;
    if (t >= N * H) return;
    const int n = t >> 2, h = t & (H - 1);
    const float* w  = Wh + (size_t)n * HF + h * OUT_F;
    const float* al = a_l + h * OUT_F;
    const float* ar = a_r + h * OUT_F;
    float sl = 0.0f, sr = 0.0f;
#pragma unroll
    for (int o = 0; o < OUT_F; ++o) {
        float v = w[o];
        sl = fmaf(v, al[o], sl);
        sr = fmaf(v, ar[o], sr);
    }
    el[t] = sl; er[t] = sr;
}

// order-preserving float atomic max (handles mixed signs; init must be -inf)
__device__ __forceinline__ void atomicMaxFloat(float* addr, float val) {
    if (val >= 0.0f) atomicMax((int*)addr, __float_as_int(val));
    else             atomicMin((unsigned int*)addr, __float_as_uint(val));
}

// ---------------------------------------------------------------------------
// Edge logits: leaky_relu(e_l[row] + e_r[col]) + log(clip(w)), segment max
// ---------------------------------------------------------------------------
__global__ void gat_edge_logit(const long long* __restrict__ ei,
                               const float* __restrict__ ew,
                               const float* __restrict__ el,
                               const float* __restrict__ er,
                               float* __restrict__ elog,
                               float* __restrict__ nmax, int E) {
    int t = blockIdx.x * blockDim.x + threadIdx.x;
    if (t >= E * H) return;
    const int e = t >> 2, h = t & (H - 1);
    const int row = (int)ei[e];
    const int col = (int)ei[(size_t)E + e];
    float v = el[row * H + h] + er[col * H + h];
    v = (v > 0.0f) ? v : NEG_SLOPE * v;
    v += __logf(fmaxf(ew[e], 1e-8f));
    elog[t] = v;
    atomicMaxFloat(&nmax[col * H + h], v);
}

// ---------------------------------------------------------------------------
// alpha_unnorm = exp(logit - max[col]); segment sum
// ---------------------------------------------------------------------------
__global__ void gat_edge_exp(const long long* __restrict__ ei,
                             float* __restrict__ elog,
                             const float* __restrict__ nmax,
                             float* __restrict__ nsum, int E) {
    int t = blockIdx.x * blockDim.x + threadIdx.x;
    if (t >= E * H) return;
    const int e = t >> 2, h = t & (H - 1);
    const int col = (int)ei[(size_t)E + e];
    float a = __expf(elog[t] - nmax[col * H + h]);
    elog[t] = a;
    atomicAdd(&nsum[col * H + h], a);
}

// ---------------------------------------------------------------------------
// Weighted scatter: one wave per (edge, head); lane = output feature (32).
// ---------------------------------------------------------------------------
__global__ void gat_scatter(const long long* __restrict__ ei,
                            const float* __restrict__ elog,
                            const float* __restrict__ nsum,
                            const float* __restrict__ Wh,
                            float* __restrict__ out, int E) {
    long long t = (long long)blockIdx.x * blockDim.x + threadIdx.x;
    const int lane = (int)(t & 31);
    const long long eh = t >> 5;                 // (e*H + h)
    if (eh >= (long long)E * H) return;
    const int e = (int)(eh >> 2), h = (int)(eh & (H - 1));
    const int row = (int)ei[e];
    const int col = (int)ei[(size_t)E + e];
    const float alpha = elog[eh] / (nsum[col * H + h] + 1e-8f);
    const float v = alpha * Wh[(size_t)row * HF + h * OUT_F + lane];
    atomicAdd(&out[(size_t)col * HF + h * OUT_F + lane], v);
}

// ---------------------------------------------------------------------------
extern "C" void kernel_launch(void* const* d_in, const int* in_sizes, int n_in,
                              void* d_out, int out_size, void* d_ws, size_t ws_size,
                              hipStream_t stream) {
    const float*     x    = (const float*)d_in[0];
    const long long* ei   = (const long long*)d_in[1];   // int64 edge_index (2,E)
    const float*     ew   = (const float*)d_in[2];
    const float*     W    = (const float*)d_in[3];
    const float*     a_l  = (const float*)d_in[4];
    const float*     a_r  = (const float*)d_in[5];
    const float*     bias = (const float*)d_in[6];
    float* out = (float*)d_out;

    const int N = in_sizes[0] / IN_F;
    const int E = in_sizes[2];

    // workspace carve-up
    char* ws = (char*)d_ws;
    float* Wh   = (float*)ws;                    ws += (size_t)N * HF * sizeof(float);
    float* el   = (float*)ws;                    ws += (size_t)N * H  * sizeof(float);
    float* er   = (float*)ws;                    ws += (size_t)N * H  * sizeof(float);
    float* nmax = (float*)ws;                    ws += (size_t)N * H  * sizeof(float);
    float* nsum = (float*)ws;                    ws += (size_t)N * H  * sizeof(float);
    float* elog = (float*)ws;                    ws += (size_t)E * H  * sizeof(float);

    const int BT = 256;

    // 1) init
    {
        int NH = N * H;
        gat_init_nodes<<<(NH + BT - 1) / BT, BT, 0, stream>>>(nmax, nsum, NH);
        int NO = N * HF;
        gat_init_out<<<(NO + BT - 1) / BT, BT, 0, stream>>>(out, bias, NO);
    }
    // 2) WMMA GEMM: Wh = x @ W
    gat_wh_gemm<<<(N + 15) / 16, BT, 0, stream>>>(x, W, Wh, N);
    // 3) attention pre-activations
    gat_elr<<<(N * H + BT - 1) / BT, BT, 0, stream>>>(Wh, a_l, a_r, el, er, N);
    // 4) edge logits + segment max
    gat_edge_logit<<<(E * H + BT - 1) / BT, BT, 0, stream>>>(ei, ew, el, er, elog, nmax, E);
    // 5) exp + segment sum
    gat_edge_exp<<<(E * H + BT - 1) / BT, BT, 0, stream>>>(ei, elog, nmax, nsum, E);
    // 6) normalize + weighted scatter-add
    {
        long long total = (long long)E * H * 32;
        long long blocks = (total + BT - 1) / BT;
        gat_scatter<<<(unsigned int)blocks, BT, 0, stream>>>(ei, elog, nsum, Wh, out, E);
    }
}